// PillarMaxPoolingV1_51015621542403
// MI455X (gfx1250) — compile-verified
//
#include <hip/hip_runtime.h>

typedef __attribute__((ext_vector_type(2))) float v2f;
typedef __attribute__((ext_vector_type(8))) float v8f;

#define LPAIRS  2000000
#define MPIL    100000
#define NTILES  (LPAIRS / 16)     // 125000 tiles of 16 rows, exact
#define CIN     10
#define CMID    32
#define COUT    64
#define BEVf    0.075f
#define PCMIN   (-54.0f)
#define BNEPS   1e-3f
#define LDS_STRIDE 34             // even (8B-aligned rows), gcd(34,64)=2 -> conflict-light

// ---------------------------------------------------------------------------
// WMMA helper: D = A(16x4 f32) * B(4x16 f32) + C(16x16 f32)
// ---------------------------------------------------------------------------
static __device__ __forceinline__ v8f wmma_f32(v2f a, v2f b, v8f c) {
  return __builtin_amdgcn_wmma_f32_16x16x4_f32(false, a, false, b, (short)0, c,
                                               false, false);
}

static __device__ __forceinline__ v8f v8zero() {
  v8f z = {0.f, 0.f, 0.f, 0.f, 0.f, 0.f, 0.f, 0.f};
  return z;
}

// Same-wave LDS store->load ordering (cross-lane through LDS).
static __device__ __forceinline__ void lds_wave_fence() {
  __builtin_amdgcn_wave_barrier();
  asm volatile("s_wait_dscnt 0" ::: "memory");
  __builtin_amdgcn_wave_barrier();
}

// ---------------------------------------------------------------------------
// B-fragment loaders.  A-layout/B-layout convention (32-bit, 16x4 / 4x16):
//   lanes 0-15  hold K = 4s+0 (reg .x) and 4s+1 (reg .y)
//   lanes 16-31 hold K = 4s+2 (reg .x) and 4s+3 (reg .y)
// B[k][j] = W[j][k]  (computing z = x @ W^T)
// ---------------------------------------------------------------------------
static __device__ __forceinline__ void load_b1(const float* __restrict__ W1,
                                               int n, int hi, v2f b1[3][2]) {
#pragma unroll
  for (int s = 0; s < 3; ++s)
#pragma unroll
    for (int t = 0; t < 2; ++t) {
      const int k = 4 * s + 2 * hi;           // even, 0..10
      v2f b;
      if (k < CIN) {                          // k=10/11 -> zero pad
        const float* p = W1 + (16 * t + n) * CIN + k;
        b.x = p[0];
        b.y = p[1];                           // k+1 <= 9 whenever k < 10
      } else {
        b.x = 0.f; b.y = 0.f;
      }
      b1[s][t] = b;
    }
}

static __device__ __forceinline__ void load_b2(const float* __restrict__ W2,
                                               int n, int hi, v2f b2[8][4]) {
#pragma unroll
  for (int s = 0; s < 8; ++s)
#pragma unroll
    for (int t = 0; t < 4; ++t)
      b2[s][t] = *(const v2f*)(W2 + (16 * t + n) * CMID + 4 * s + 2 * hi);
}

// ---------------------------------------------------------------------------
// Layer 1 for one 16-row tile: gather features, 6x WMMA -> z1 (pre-BN)
// c1[t] holds channels 16t..16t+15 ; lane (n,hi) VGPR r = row (r + 8*hi)
// ---------------------------------------------------------------------------
static __device__ __forceinline__ void tile_z1(
    int base, int n, int hi,
    const float* __restrict__ xyz, const float* __restrict__ pfeat,
    const int* __restrict__ gpt, const int* __restrict__ gpl,
    const int* __restrict__ pind, const v2f b1[3][2], v8f c1[2]) {
  const int l  = base + n;
  const int pt = gpt[l];
  const float* pfr = pfeat + pt * 8;
  v2f a[3];
  if (hi == 0) {
    const int p  = gpl[l];
    const float cx = ((float)pind[p * 3 + 2] + 0.5f) * BEVf + PCMIN;
    const float cy = ((float)pind[p * 3 + 1] + 0.5f) * BEVf + PCMIN;
    a[0].x = xyz[pt * 3 + 0] - cx;            // k=0
    a[0].y = xyz[pt * 3 + 1] - cy;            // k=1
    a[1].x = pfr[2]; a[1].y = pfr[3];         // k=4,5
    a[2].x = pfr[6]; a[2].y = pfr[7];         // k=8,9
  } else {
    a[0].x = pfr[0]; a[0].y = pfr[1];         // k=2,3
    a[1].x = pfr[4]; a[1].y = pfr[5];         // k=6,7
    a[2].x = 0.f;    a[2].y = 0.f;            // k=10,11 zero pad
  }
  c1[0] = v8zero();
  c1[1] = v8zero();
#pragma unroll
  for (int s = 0; s < 3; ++s) {
    c1[0] = wmma_f32(a[s], b1[s][0], c1[0]);
    c1[1] = wmma_f32(a[s], b1[s][1], c1[1]);
  }
}

// BN+ReLU layer 1 and transpose-store tile into per-wave LDS (16 rows x 32 ch)
static __device__ __forceinline__ void store_h1(float* ldsT, int n, int hi,
                                                const v8f c1[2],
                                                const float* __restrict__ sc1,
                                                const float* __restrict__ sh1) {
#pragma unroll
  for (int t = 0; t < 2; ++t) {
    const float sc = sc1[16 * t + n];
    const float sh = sh1[16 * t + n];
#pragma unroll
    for (int r = 0; r < 8; ++r) {
      const float v = fmaxf(c1[t][r] * sc + sh, 0.f);
      ldsT[(r + 8 * hi) * LDS_STRIDE + 16 * t + n] = v;
    }
  }
}

// Layer 2: read A fragments from LDS (row=tile row, K=32 channels), 32x WMMA
static __device__ __forceinline__ void tile_z2(const float* ldsT, int n, int hi,
                                               const v2f b2[8][4], v8f c2[4]) {
#pragma unroll
  for (int t = 0; t < 4; ++t) c2[t] = v8zero();
#pragma unroll
  for (int s = 0; s < 8; ++s) {
    v2f a = *(const v2f*)(ldsT + n * LDS_STRIDE + 4 * s + 2 * hi);
#pragma unroll
    for (int t = 0; t < 4; ++t) c2[t] = wmma_f32(a, b2[s][t], c2[t]);
  }
}

// ---------------------------------------------------------------------------
// Pass 1: per-channel sum / sumsq of z1 over all L rows
// ---------------------------------------------------------------------------
__global__ void __launch_bounds__(256) pass1_stats(
    const float* __restrict__ xyz, const float* __restrict__ pfeat,
    const int* __restrict__ pind, const int* __restrict__ gpt,
    const int* __restrict__ gpl, const float* __restrict__ W1,
    float* __restrict__ gstats1) {
  __shared__ float sstat[64];
  const int tid = threadIdx.x;
  if (tid < 64) sstat[tid] = 0.f;
  __syncthreads();

  const int lane = tid & 31, n = lane & 15, hi = lane >> 4;
  const int gwave  = blockIdx.x * (blockDim.x >> 5) + (tid >> 5);
  const int nwaves = gridDim.x * (blockDim.x >> 5);

  v2f b1[3][2];
  load_b1(W1, n, hi, b1);

  float s[2] = {0.f, 0.f}, sq[2] = {0.f, 0.f};
  for (int tile = gwave; tile < NTILES; tile += nwaves) {
    v8f c1[2];
    tile_z1(tile * 16, n, hi, xyz, pfeat, gpt, gpl, pind, b1, c1);
#pragma unroll
    for (int t = 0; t < 2; ++t)
#pragma unroll
      for (int r = 0; r < 8; ++r) {
        const float v = c1[t][r];
        s[t] += v; sq[t] += v * v;
      }
  }
#pragma unroll
  for (int t = 0; t < 2; ++t) {
    atomicAdd(&sstat[16 * t + n], s[t]);
    atomicAdd(&sstat[32 + 16 * t + n], sq[t]);
  }
  __syncthreads();
  if (tid < 64) atomicAdd(&gstats1[tid], sstat[tid]);
}

// ---------------------------------------------------------------------------
// BN finalize: scale = gamma*rsqrt(var+eps); shift = beta - mu*scale
// ---------------------------------------------------------------------------
__global__ void finalize_bn(const float* __restrict__ gstats,
                            const float* __restrict__ gamma,
                            const float* __restrict__ beta,
                            float* __restrict__ scsh, int nch) {
  const int i = threadIdx.x;
  if (i < nch) {
    const float invL = 1.0f / (float)LPAIRS;
    const float mu  = gstats[i] * invL;
    const float var = gstats[nch + i] * invL - mu * mu;
    const float sc  = gamma[i] * rsqrtf(var + BNEPS);
    scsh[i]       = sc;
    scsh[nch + i] = beta[i] - mu * sc;
  }
}

// ---------------------------------------------------------------------------
// Pass 2: recompute z1 -> BN+ReLU -> z2, per-channel sum/sumsq of z2 (64 ch)
// ---------------------------------------------------------------------------
__global__ void __launch_bounds__(256) pass2_stats(
    const float* __restrict__ xyz, const float* __restrict__ pfeat,
    const int* __restrict__ pind, const int* __restrict__ gpt,
    const int* __restrict__ gpl, const float* __restrict__ W1,
    const float* __restrict__ W2, const float* __restrict__ scsh1,
    float* __restrict__ gstats2) {
  __shared__ float sstat[128];
  __shared__ __align__(16) float ldsT[8][16 * LDS_STRIDE];
  const int tid = threadIdx.x;
  if (tid < 128) sstat[tid] = 0.f;
  __syncthreads();

  const int lane = tid & 31, n = lane & 15, hi = lane >> 4;
  const int wib = tid >> 5;
  const int gwave  = blockIdx.x * (blockDim.x >> 5) + wib;
  const int nwaves = gridDim.x * (blockDim.x >> 5);
  float* myT = &ldsT[wib][0];

  v2f b1[3][2]; load_b1(W1, n, hi, b1);
  v2f b2[8][4]; load_b2(W2, n, hi, b2);
  const float* sc1 = scsh1;
  const float* sh1 = scsh1 + 32;

  float s[4] = {0.f, 0.f, 0.f, 0.f}, sq[4] = {0.f, 0.f, 0.f, 0.f};
  for (int tile = gwave; tile < NTILES; tile += nwaves) {
    v8f c1[2];
    tile_z1(tile * 16, n, hi, xyz, pfeat, gpt, gpl, pind, b1, c1);
    store_h1(myT, n, hi, c1, sc1, sh1);
    lds_wave_fence();
    v8f c2[4];
    tile_z2(myT, n, hi, b2, c2);
#pragma unroll
    for (int t = 0; t < 4; ++t)
#pragma unroll
      for (int r = 0; r < 8; ++r) {
        const float v = c2[t][r];
        s[t] += v; sq[t] += v * v;
      }
  }
#pragma unroll
  for (int t = 0; t < 4; ++t) {
    atomicAdd(&sstat[16 * t + n], s[t]);
    atomicAdd(&sstat[64 + 16 * t + n], sq[t]);
  }
  __syncthreads();
  if (tid < 128) atomicAdd(&gstats2[tid], sstat[tid]);
}

// ---------------------------------------------------------------------------
// Pass 3: full recompute + BN2 + ReLU + scatter-max to pillars
// h >= 0 and out zero-initialized -> uint atomicMax == float max (ref match)
// ---------------------------------------------------------------------------
__global__ void __launch_bounds__(256) pass3_scatter(
    const float* __restrict__ xyz, const float* __restrict__ pfeat,
    const int* __restrict__ pind, const int* __restrict__ gpt,
    const int* __restrict__ gpl, const float* __restrict__ W1,
    const float* __restrict__ W2, const float* __restrict__ scsh1,
    const float* __restrict__ scsh2, float* __restrict__ out) {
  __shared__ __align__(16) float ldsT[8][16 * LDS_STRIDE];
  const int tid = threadIdx.x;
  const int lane = tid & 31, n = lane & 15, hi = lane >> 4;
  const int wib = tid >> 5;
  const int gwave  = blockIdx.x * (blockDim.x >> 5) + wib;
  const int nwaves = gridDim.x * (blockDim.x >> 5);
  float* myT = &ldsT[wib][0];

  v2f b1[3][2]; load_b1(W1, n, hi, b1);
  v2f b2[8][4]; load_b2(W2, n, hi, b2);
  const float* sc1 = scsh1;
  const float* sh1 = scsh1 + 32;
  const float* sc2 = scsh2;
  const float* sh2 = scsh2 + 64;
  unsigned* outu = (unsigned*)out;

  for (int tile = gwave; tile < NTILES; tile += nwaves) {
    const int base = tile * 16;
    v8f c1[2];
    tile_z1(base, n, hi, xyz, pfeat, gpt, gpl, pind, b1, c1);
    store_h1(myT, n, hi, c1, sc1, sh1);
    lds_wave_fence();
    v8f c2[4];
    tile_z2(myT, n, hi, b2, c2);
#pragma unroll
    for (int r = 0; r < 8; ++r) {
      const int l = base + r + 8 * hi;
      const int p = gpl[l];                    // sorted -> L2-friendly
#pragma unroll
      for (int t = 0; t < 4; ++t) {
        const float v = fmaxf(c2[t][r] * sc2[16 * t + n] + sh2[16 * t + n], 0.f);
        atomicMax(&outu[p * COUT + 16 * t + n], __float_as_uint(v));
      }
    }
  }
}

// ---------------------------------------------------------------------------
// Zero output (poisoned by harness, atomics need clean base) + stats scratch
// ---------------------------------------------------------------------------
__global__ void zero_kernel(float* __restrict__ out, float* __restrict__ ws,
                            int nout) {
  const int stride = gridDim.x * blockDim.x;
  int i = blockIdx.x * blockDim.x + threadIdx.x;
  for (int j = i; j < nout; j += stride) out[j] = 0.f;
  if (i < 384) ws[i] = 0.f;
}

// ---------------------------------------------------------------------------
// Host launch.  ws float layout: [0,64)=stats1  [64,128)=scsh1
//                               [128,256)=stats2 [256,384)=scsh2
// ---------------------------------------------------------------------------
extern "C" void kernel_launch(void* const* d_in, const int* in_sizes, int n_in,
                              void* d_out, int out_size, void* d_ws,
                              size_t ws_size, hipStream_t stream) {
  const float* xyz    = (const float*)d_in[0];
  const float* pfeat  = (const float*)d_in[1];
  // d_in[2] = xyz_batch_cnt (unused by the math)
  const int*   pind   = (const int*)d_in[3];
  const int*   gpt    = (const int*)d_in[4];
  const int*   gpl    = (const int*)d_in[5];
  const float* W1     = (const float*)d_in[6];
  const float* gamma1 = (const float*)d_in[7];
  const float* beta1  = (const float*)d_in[8];
  const float* W2     = (const float*)d_in[9];
  const float* gamma2 = (const float*)d_in[10];
  const float* beta2  = (const float*)d_in[11];
  float* out = (float*)d_out;

  float* ws     = (float*)d_ws;
  float* stats1 = ws;
  float* scsh1  = ws + 64;
  float* stats2 = ws + 128;
  float* scsh2  = ws + 256;

  const int blocks = 1024, threads = 256;

  zero_kernel<<<blocks, threads, 0, stream>>>(out, ws, MPIL * COUT);
  pass1_stats<<<blocks, threads, 0, stream>>>(xyz, pfeat, pind, gpt, gpl, W1,
                                              stats1);
  finalize_bn<<<1, 64, 0, stream>>>(stats1, gamma1, beta1, scsh1, CMID);
  pass2_stats<<<blocks, threads, 0, stream>>>(xyz, pfeat, pind, gpt, gpl, W1,
                                              W2, scsh1, stats2);
  finalize_bn<<<1, 64, 0, stream>>>(stats2, gamma2, beta2, scsh2, COUT);
  pass3_scatter<<<blocks, threads, 0, stream>>>(xyz, pfeat, pind, gpt, gpl, W1,
                                                W2, scsh1, scsh2, out);
}